// ResamplerLayer_15668040696111
// MI455X (gfx1250) — compile-verified
//
#include <hip/hip_runtime.h>

// Trilinear resampling, REPLICATE boundary.
// inputs:        [B, X, Y, Z, C] f32   (B=2, X=Y=Z=128, C=4)
// sample_coords: [B, D, D, D, 3] f32   (D=96), voxel-space coords in [-1, size]
// out:           [B, D, D, D, C] f32

typedef float f32x4 __attribute__((ext_vector_type(4)));

constexpr int KB  = 2;     // batch
constexpr int KX  = 128;   // input spatial dims
constexpr int KY  = 128;
constexpr int KZ  = 128;
constexpr int KC  = 4;     // channels (one float4 / b128 per voxel)
constexpr int KD  = 96;    // output grid per dim
constexpr int NSAMP = KB * KD * KD * KD;            // 1,769,472 samples
constexpr int SAMP_PER_BATCH = KD * KD * KD;        // 884,736

static __device__ __forceinline__ int clampi(int v, int lo, int hi) {
    v = v < lo ? lo : v;
    return v > hi ? hi : v;
}

__global__ __launch_bounds__(256) void resampler_trilinear_kernel(
    const float* __restrict__ in,       // [B,X,Y,Z,C]
    const float* __restrict__ coords,   // [B,D,D,D,3]
    f32x4* __restrict__ out)            // [B,D,D,D] of float4
{
    const int i = blockIdx.x * 256 + threadIdx.x;
    if (i >= NSAMP) return;

    // ---- coords: streamed once -> non-temporal loads (don't pollute L2,
    // which we want to keep holding the 64MB input volume). -----------------
    const float* cp = coords + 3l * i;
    const float cx = __builtin_nontemporal_load(cp + 0);
    const float cy = __builtin_nontemporal_load(cp + 1);
    const float cz = __builtin_nontemporal_load(cp + 2);

    // Warm L2 for coords ~1MB ahead of this block (speculative prefetch;
    // dropped by HW if translation fails, but guard anyway).
    const long pf = 3l * i + 3l * 98304;
    if (pf + 2 < 3l * NSAMP)
        __builtin_prefetch(coords + pf, 0, 1);   // -> global_prefetch_b8

    // ---- weights + clamped corner indices (REPLICATE) ---------------------
    const float fx = floorf(cx), fy = floorf(cy), fz = floorf(cz);
    const float wx = cx - fx,    wy = cy - fy,    wz = cz - fz;
    const int x0 = (int)fx, y0 = (int)fy, z0 = (int)fz;

    const int x0c = clampi(x0,     0, KX - 1);
    const int x1c = clampi(x0 + 1, 0, KX - 1);
    const int y0c = clampi(y0,     0, KY - 1);
    const int y1c = clampi(y0 + 1, 0, KY - 1);
    const int z0c = clampi(z0,     0, KZ - 1);
    const int z1c = clampi(z0 + 1, 0, KZ - 1);

    // ---- 8 corner gathers: one global_load_b128 each (C=4 floats) ---------
    const int b = i / SAMP_PER_BATCH;
    const f32x4* __restrict__ base =
        (const f32x4*)in + (long)b * ((long)KX * KY * KZ);

    const long r00 = ((long)x0c * KY + y0c) * KZ;
    const long r01 = ((long)x0c * KY + y1c) * KZ;
    const long r10 = ((long)x1c * KY + y0c) * KZ;
    const long r11 = ((long)x1c * KY + y1c) * KZ;

    const f32x4 v000 = base[r00 + z0c];
    const f32x4 v001 = base[r00 + z1c];
    const f32x4 v010 = base[r01 + z0c];
    const f32x4 v011 = base[r01 + z1c];
    const f32x4 v100 = base[r10 + z0c];
    const f32x4 v101 = base[r10 + z1c];
    const f32x4 v110 = base[r11 + z0c];
    const f32x4 v111 = base[r11 + z1c];

    // ---- factored trilinear blend: 7 lerps (sub + fma per channel) --------
    const f32x4 c00 = v000 + wz * (v001 - v000);
    const f32x4 c01 = v010 + wz * (v011 - v010);
    const f32x4 c10 = v100 + wz * (v101 - v100);
    const f32x4 c11 = v110 + wz * (v111 - v110);

    const f32x4 c0 = c00 + wy * (c01 - c00);
    const f32x4 c1 = c10 + wy * (c11 - c10);

    const f32x4 r = c0 + wx * (c1 - c0);

    // ---- output: written once -> non-temporal b128 store ------------------
    __builtin_nontemporal_store(r, out + i);
}

extern "C" void kernel_launch(void* const* d_in, const int* in_sizes, int n_in,
                              void* d_out, int out_size, void* d_ws, size_t ws_size,
                              hipStream_t stream) {
    (void)in_sizes; (void)n_in; (void)out_size; (void)d_ws; (void)ws_size;
    const float* inputs = (const float*)d_in[0];
    const float* coords = (const float*)d_in[1];
    f32x4* out = (f32x4*)d_out;

    const int blocks = (NSAMP + 255) / 256;   // 6912 blocks of 8 wave32s
    resampler_trilinear_kernel<<<blocks, 256, 0, stream>>>(inputs, coords, out);
}